// UpSampleInterpolation_90297392431680
// MI455X (gfx1250) — compile-verified
//
#include <hip/hip_runtime.h>
#include <hip/hip_bf16.h>

// ---------------------------------------------------------------------------
// UpSampleInterpolation for MI455X (gfx1250, wave32, WMMA).
// Sizes fixed by reference: B=4, N=16384, S=4096, Cd=128, Cs=256, Co=128, K=3
// ---------------------------------------------------------------------------

typedef float v2f __attribute__((ext_vector_type(2)));
typedef float v4f __attribute__((ext_vector_type(4)));
typedef float v8f __attribute__((ext_vector_type(8)));

#define NB   4
#define NN   16384
#define NS   4096
#define CD   128
#define CS   256
#define CO   128
#define CIN  (CD + CS)      // 384
#define EPSF 1e-8f
#define BN_EPSF 1e-5f
#define NEG_SLOPE 0.2f

// ---------------------------------------------------------------------------
// Kernel 1: KNN via V_WMMA_F32_16X16X4_F32.
// One wave handles 16 dense points (columns). Each WMMA produces a
// 16(sparse)x16(dense) tile of squared distances:
//   A[m]   = (sx, sy, sz, |s|^2)      (sparse point m of the tile)
//   B[:,n] = (-2dx, -2dy, -2dz, 1)
//   C[m][n] = |d_n|^2
//   => D[m][n] = |s|^2 + |d|^2 - 2 s.d
// D layout: lane L (<16) holds rows m=0..7 for column n=L; lane L+16 holds
// rows m=8..15 for the same column => top-3 over S is lane-local; lanes L
// and L+16 each scan half of S and merge once at the end via shfl_xor(16).
// ---------------------------------------------------------------------------
__global__ __launch_bounds__(256)
void knn_idw_kernel(const float* __restrict__ dxyz,   // (B,3,N)
                    const float* __restrict__ sxyz,   // (B,3,S)
                    int*   __restrict__ outIdx,       // (B,N,3)
                    float* __restrict__ outW)         // (B,N,3) normalized
{
    __shared__ float ls[NS * 4];   // per sparse point: {x, y, z, |s|^2} = 64KB

    const int tid = threadIdx.x;
    const int b = blockIdx.x >> 7;          // 128 blocks per batch
    const int blockInB = blockIdx.x & 127;

    // Stage sparse points of this batch into LDS (coalesced global reads).
    const float* sb = sxyz + (size_t)b * 3 * NS;
    for (int s = tid; s < NS; s += 256) {
        float x = sb[s], y = sb[NS + s], z = sb[2 * NS + s];
        ls[4 * s + 0] = x;
        ls[4 * s + 1] = y;
        ls[4 * s + 2] = z;
        ls[4 * s + 3] = x * x + y * y + z * z;
    }
    __syncthreads();

    const int wave = tid >> 5;
    const int lane = tid & 31;
    const int col  = lane & 15;
    const bool hi  = lane >= 16;

    const int nbase = (blockInB * 8 + wave) * 16;
    const int n = nbase + col;

    // B operand (dense tile), constant over the sparse loop.
    const float* db = dxyz + (size_t)b * 3 * NN;
    const float px = db[n], py = db[NN + n], pz = db[2 * NN + n];
    const float n2 = px * px + py * py + pz * pz;
    v2f bmat;
    bmat.x = hi ? (-2.0f * pz) : (-2.0f * px);   // k=2 : k=0
    bmat.y = hi ? 1.0f         : (-2.0f * py);   // k=3 : k=1

    v8f cmat;
#pragma unroll
    for (int r = 0; r < 8; ++r) cmat[r] = n2;

    float d0 = 1e30f, d1 = 1e30f, d2v = 1e30f;
    int   i0 = 0,     i1 = 0,     i2 = 0;

#pragma unroll 2
    for (int t = 0; t < NS / 16; ++t) {
        const int s = t * 16 + col;
        // (x,y) for lanes 0-15 (k=0,1); (z,|s|^2) for lanes 16-31 (k=2,3)
        const v2f av = *(const v2f*)(ls + 4 * s + (hi ? 2 : 0));

        v8f d = __builtin_amdgcn_wmma_f32_16x16x4_f32(
            false, av, false, bmat, (short)0, cmat, false, false);

        const int sbase = t * 16 + (hi ? 8 : 0);
#pragma unroll
        for (int r = 0; r < 8; ++r) {
            float dist = fmaxf(d[r], 0.0f);
            int   si   = sbase + r;
            bool lt2 = dist < d2v;
            bool lt1 = dist < d1;
            bool lt0 = dist < d0;
            float nd2 = lt1 ? d1 : (lt2 ? dist : d2v);
            int   ni2 = lt1 ? i1 : (lt2 ? si   : i2);
            float nd1 = lt0 ? d0 : (lt1 ? dist : d1);
            int   ni1 = lt0 ? i0 : (lt1 ? si   : i1);
            float nd0 = lt0 ? dist : d0;
            int   ni0 = lt0 ? si   : i0;
            d0 = nd0; d1 = nd1; d2v = nd2;
            i0 = ni0; i1 = ni1; i2  = ni2;
        }
    }

    // Merge with partner lane (lane ^ 16): same dense column, other half of S.
    float pd0 = __shfl_xor(d0, 16), pd1 = __shfl_xor(d1, 16), pd2 = __shfl_xor(d2v, 16);
    int   pi0 = __shfl_xor(i0, 16), pi1 = __shfl_xor(i1, 16), pi2 = __shfl_xor(i2, 16);

#pragma unroll
    for (int j = 0; j < 3; ++j) {
        float dist = (j == 0) ? pd0 : (j == 1) ? pd1 : pd2;
        int   si   = (j == 0) ? pi0 : (j == 1) ? pi1 : pi2;
        bool lt2 = dist < d2v;
        bool lt1 = dist < d1;
        bool lt0 = dist < d0;
        float nd2 = lt1 ? d1 : (lt2 ? dist : d2v);
        int   ni2 = lt1 ? i1 : (lt2 ? si   : i2);
        float nd1 = lt0 ? d0 : (lt1 ? dist : d1);
        int   ni1 = lt0 ? i0 : (lt1 ? si   : i1);
        float nd0 = lt0 ? dist : d0;
        int   ni0 = lt0 ? si   : i0;
        d0 = nd0; d1 = nd1; d2v = nd2;
        i0 = ni0; i1 = ni1; i2  = ni2;
    }

    if (!hi) {
        float w0 = 1.0f / (d0  + EPSF);
        float w1 = 1.0f / (d1  + EPSF);
        float w2 = 1.0f / (d2v + EPSF);
        float inv = 1.0f / (w0 + w1 + w2);
        size_t g = ((size_t)b * NN + n) * 3;
        outIdx[g + 0] = i0; outIdx[g + 1] = i1; outIdx[g + 2] = i2;
        outW[g + 0] = w0 * inv; outW[g + 1] = w1 * inv; outW[g + 2] = w2 * inv;
    }
}

// ---------------------------------------------------------------------------
// Kernel 2: fused interpolation + concat + 1x1 conv (GEMM) -> pre-BN y.
// One block = one (b, 32-column n-tile), stored as two 384x16 sub-tiles in
// LDS (48KB). Each of the 8 waves owns 16 co rows and issues two chained
// V_WMMA_F32_16X16X4_F32 per K-step that SHARE the A (weight) registers,
// halving W loads and per-WMMA address math. 96 K-steps (K=384, 4/step).
// ---------------------------------------------------------------------------
__global__ __launch_bounds__(256)
void interp_gemm_kernel(const float* __restrict__ ddata,  // (B,Cd,N)
                        const float* __restrict__ sdata,  // (B,Cs,S)
                        const float* __restrict__ Wm,     // (Co, 384) row-major
                        const int*   __restrict__ knnI,
                        const float* __restrict__ knnW,
                        float* __restrict__ ypre)         // (B,Co,N)
{
    __shared__ float xs[2 * CIN * 16];   // [sub][c][n16], 48KB
    __shared__ int   lidx[32 * 3];
    __shared__ float lw[32 * 3];

    const int tid = threadIdx.x;
    const int b  = blockIdx.x >> 9;      // N/32 = 512 tiles per batch
    const int nb = blockIdx.x & 511;
    const int n0 = nb * 32;

    if (tid < 96) {
        size_t g = ((size_t)b * NN + n0) * 3 + tid;
        lidx[tid] = knnI[g];
        lw[tid]   = knnW[g];
    }

    // Dense rows 0..127 of both sub-tiles (coalesced 128B rows).
    const float* dbp = ddata + (size_t)b * CD * NN;
    for (int i = tid; i < CD * 32; i += 256) {
        int c = i >> 5, j = i & 31;                 // j = column within 32
        int sub = j >> 4, jj = j & 15;
        xs[(sub * CIN + c) * 16 + jj] = dbp[(size_t)c * NN + n0 + j];
    }
    __syncthreads();

    // Interp rows 128..383: thread tid = channel c; gathers hit L2 (16MB set).
    {
        const float* row = sdata + ((size_t)b * CS + tid) * NS;
        for (int j = 0; j < 32; ++j) {
            int   g0 = lidx[j * 3 + 0], g1 = lidx[j * 3 + 1], g2 = lidx[j * 3 + 2];
            float w0 = lw[j * 3 + 0],   w1 = lw[j * 3 + 1],   w2 = lw[j * 3 + 2];
            int sub = j >> 4, jj = j & 15;
            xs[(sub * CIN + CD + tid) * 16 + jj] =
                w0 * row[g0] + w1 * row[g1] + w2 * row[g2];
        }
    }
    __syncthreads();

    // GEMM: wave w -> co rows [16w, 16w+16), both 16-column sub-tiles.
    const int wave = tid >> 5;
    const int lane = tid & 31;
    const int col  = lane & 15;
    const bool hi  = lane >= 16;
    const int cobase = wave * 16;

    const float* wrow = Wm + (size_t)(cobase + col) * CIN + (hi ? 2 : 0);

    v8f accA = {};
    v8f accB = {};
#pragma unroll 4
    for (int k = 0; k < CIN; k += 4) {
        v2f a = *(const v2f*)(wrow + k);           // W[m][k..k+1] (or k+2..k+3)
        const int kk = k + (hi ? 2 : 0);
        v2f bvA, bvB;
        bvA.x = xs[kk * 16 + col];
        bvA.y = xs[(kk + 1) * 16 + col];
        bvB.x = xs[(CIN + kk) * 16 + col];
        bvB.y = xs[(CIN + kk + 1) * 16 + col];
        accA = __builtin_amdgcn_wmma_f32_16x16x4_f32(
            false, a, false, bvA, (short)0, accA, false, false);
        accB = __builtin_amdgcn_wmma_f32_16x16x4_f32(
            false, a, false, bvB, (short)0, accB, false, false);
    }

    float* yb = ypre + (size_t)b * CO * NN;
    const int rowbase = cobase + (hi ? 8 : 0);
#pragma unroll
    for (int r = 0; r < 8; ++r) {
        yb[(size_t)(rowbase + r) * NN + n0 + col]      = accA[r];
        yb[(size_t)(rowbase + r) * NN + n0 + 16 + col] = accB[r];
    }
}

// ---------------------------------------------------------------------------
// Kernel 3a/3b: deterministic two-stage BN statistics (no float atomics).
// ---------------------------------------------------------------------------
__global__ __launch_bounds__(256)
void bn_partial_kernel(const float* __restrict__ y, float* __restrict__ partial)
{
    const int co = blockIdx.x >> 4;   // 128 channels
    const int ch = blockIdx.x & 15;   // 16 chunks of 4096 elems
    const int tid = threadIdx.x;

    float s = 0.0f, ss = 0.0f;
    // 4096 elems per block, float4 loads: 1024 vectors, 4 per thread.
    for (int ev = ch * 1024 + tid; ev < ch * 1024 + 1024; ev += 256) {
        int e = ev * 4;
        int b = e >> 14, n = e & 16383;
        v4f v = *(const v4f*)(y + ((size_t)b * CO + co) * NN + n);
        s  += v.x + v.y + v.z + v.w;
        ss += v.x * v.x + v.y * v.y + v.z * v.z + v.w * v.w;
    }

    __shared__ float rs[256], rss[256];
    rs[tid] = s; rss[tid] = ss;
    __syncthreads();
    for (int off = 128; off > 0; off >>= 1) {
        if (tid < off) { rs[tid] += rs[tid + off]; rss[tid] += rss[tid + off]; }
        __syncthreads();
    }
    if (tid == 0) {
        partial[(co * 16 + ch) * 2 + 0] = rs[0];
        partial[(co * 16 + ch) * 2 + 1] = rss[0];
    }
}

__global__ void bn_finalize_kernel(const float* __restrict__ partial,
                                   const float* __restrict__ gamma,
                                   const float* __restrict__ beta,
                                   float* __restrict__ scale_bias)
{
    const int co = threadIdx.x;  // blockDim = 128
    float s = 0.0f, ss = 0.0f;
    for (int ch = 0; ch < 16; ++ch) {
        s  += partial[(co * 16 + ch) * 2 + 0];
        ss += partial[(co * 16 + ch) * 2 + 1];
    }
    const float invN = 1.0f / 65536.0f;
    float mean = s * invN;
    float var  = ss * invN - mean * mean;
    float istd = rsqrtf(var + BN_EPSF);
    float sc = gamma[co] * istd;
    scale_bias[co * 2 + 0] = sc;
    scale_bias[co * 2 + 1] = beta[co] - mean * sc;
}

// ---------------------------------------------------------------------------
// Kernel 4: in-place BN affine + LeakyReLU (float4).
// ---------------------------------------------------------------------------
__global__ __launch_bounds__(256)
void bn_apply_kernel(float* __restrict__ y, const float* __restrict__ scale_bias)
{
    size_t iv = (size_t)blockIdx.x * 256 + threadIdx.x;   // float4 index
    size_t i = iv * 4;
    int co = (int)((i >> 14) & 127);                      // NN%4==0 => uniform
    float sc = scale_bias[co * 2], bi = scale_bias[co * 2 + 1];
    v4f v = *(v4f*)(y + i);
    v.x = v.x * sc + bi; v.y = v.y * sc + bi;
    v.z = v.z * sc + bi; v.w = v.w * sc + bi;
    v.x = (v.x >= 0.0f) ? v.x : NEG_SLOPE * v.x;
    v.y = (v.y >= 0.0f) ? v.y : NEG_SLOPE * v.y;
    v.z = (v.z >= 0.0f) ? v.z : NEG_SLOPE * v.z;
    v.w = (v.w >= 0.0f) ? v.w : NEG_SLOPE * v.w;
    *(v4f*)(y + i) = v;
}

__global__ __launch_bounds__(256)
void copy_xyz_kernel(const float* __restrict__ src, float* __restrict__ dst)
{
    int i = blockIdx.x * 256 + threadIdx.x;               // float4 index
    *(v4f*)(dst + (size_t)i * 4) = *(const v4f*)(src + (size_t)i * 4);
}

// ---------------------------------------------------------------------------
extern "C" void kernel_launch(void* const* d_in, const int* in_sizes, int n_in,
                              void* d_out, int out_size, void* d_ws, size_t ws_size,
                              hipStream_t stream)
{
    const float* dxyz  = (const float*)d_in[0];  // (B,3,N)
    const float* sxyz  = (const float*)d_in[1];  // (B,3,S)
    const float* ddata = (const float*)d_in[2];  // (B,Cd,N)
    const float* sdata = (const float*)d_in[3];  // (B,Cs,S)
    const float* Wm    = (const float*)d_in[4];  // (Co, Cd+Cs)
    const float* gamma = (const float*)d_in[5];  // (Co)
    const float* beta  = (const float*)d_in[6];  // (Co)

    float* y = (float*)d_out;                                 // (B,Co,N)
    float* xyz_out = y + (size_t)NB * CO * NN;                // tuple elem 2

    // Workspace layout (256B-aligned offsets).
    char* ws = (char*)d_ws;
    int*   knnI    = (int*)(ws + 0);                 // 786432 B
    float* knnW    = (float*)(ws + 786432);          // 786432 B
    float* partial = (float*)(ws + 1572864);         // 16384 B
    float* sclb    = (float*)(ws + 1589248);         // 1024 B

    // 1) KNN + inverse-distance weights (WMMA f32 16x16x4 distance tiles).
    knn_idw_kernel<<<NB * 128, 256, 0, stream>>>(dxyz, sxyz, knnI, knnW);

    // 2) interp + concat + conv (WMMA f32 GEMM, dual-tile A reuse).
    interp_gemm_kernel<<<NB * (NN / 32), 256, 0, stream>>>(
        ddata, sdata, Wm, knnI, knnW, y);

    // 3) BN statistics (deterministic two-stage reduction).
    bn_partial_kernel<<<CO * 16, 256, 0, stream>>>(y, partial);
    bn_finalize_kernel<<<1, 128, 0, stream>>>(partial, gamma, beta, sclb);

    // 4) BN affine + LeakyReLU in place; copy xyz passthrough output.
    bn_apply_kernel<<<(NB * CO * NN) / 1024, 256, 0, stream>>>(y, sclb);
    copy_xyz_kernel<<<(NB * 3 * NN) / 1024, 256, 0, stream>>>(dxyz, xyz_out);
}